// BasicConvolutionBlock_37452114821414
// MI455X (gfx1250) — compile-verified
//
#include <hip/hip_runtime.h>
#include <hip/hip_bf16.h>

#define N_VOX   200000
#define CIN     64
#define COUT    64
#define KOFF    27
#define BN_EPS  1e-5f

#define ROWS_PER_WAVE   32    // two 16-row WMMA tiles per wave
#define WAVES_PER_BLOCK 8
#define ROWS_PER_BLOCK  256   // 8 waves * 32 rows
#define WPAD            68    // LDS row stride (floats): 64 + 4 pad -> no lane/lane+16 bank conflict

typedef float v2f __attribute__((ext_vector_type(2)));
typedef float v8f __attribute__((ext_vector_type(8)));

// ---------------------------------------------------------------------------
// Kernel 1: gather + 27x GEMM accumulate via V_WMMA_F32_16X16X4_F32.
// Each wave owns a 32x64 output tile (two 16x64 WMMA tiles) so every B
// fragment read from LDS feeds two WMMAs. Writes conv[N,64] and per-block
// column sum/sumsq partials for the BN reduction.
// ---------------------------------------------------------------------------
__global__ __launch_bounds__(256) void conv_wmma_kernel(
    const float* __restrict__ feats,      // [N, 64]
    const float* __restrict__ W,          // [27, 64, 64]
    const int*   __restrict__ nbr_idx,    // [27, N]
    const int*   __restrict__ mask,       // [27, N]
    float*       __restrict__ conv_out,   // [N, 64]
    float*       __restrict__ part)       // [gridDim.x, 128] (64 sum + 64 sumsq)
{
    __shared__ float Wlds[CIN * WPAD];               // 17408 B
    __shared__ float red[WAVES_PER_BLOCK * 128];     //  4096 B

    const int tid   = threadIdx.x;
    const int lane  = tid & 31;
    const int wave  = tid >> 5;
    const int lrow  = lane & 15;    // row-in-tile / col-in-tile index
    const int lhalf = lane >> 4;    // 0: lanes 0-15, 1: lanes 16-31
    const int row0  = blockIdx.x * ROWS_PER_BLOCK + wave * ROWS_PER_WAVE;
    const bool valid = (row0 + ROWS_PER_WAVE) <= N_VOX;   // N is a multiple of 32

    // acc[tile][ctile]: tile 0 = rows row0..row0+15, tile 1 = rows row0+16..row0+31
    v8f a00 = {}, a01 = {}, a02 = {}, a03 = {};
    v8f a10 = {}, a11 = {}, a12 = {}, a13 = {};

    for (int k = 0; k < KOFF; ++k) {
        // ---- stage W_k (64x64 fp32 = 16 KB) into padded LDS ----
        __syncthreads();   // previous iteration's LDS reads done
        {
            const float4* Wg = (const float4*)(W + (size_t)k * CIN * COUT);
            #pragma unroll
            for (int j = 0; j < 4; ++j) {
                int e4  = tid + j * 256;       // float4 index, coalesced
                int e   = e4 * 4;
                int row = e >> 6;
                int col = e & 63;
                *(float4*)(&Wlds[row * WPAD + col]) = Wg[e4];
            }
        }
        __syncthreads();

        // ---- per-lane gather pointers for the wave's two 16-row tiles ----
        int  g0 = 0, g1 = 0;
        bool m0 = false, m1 = false;
        if (valid) {
            int r0 = row0 + lrow;
            int r1 = row0 + 16 + lrow;
            g0 = nbr_idx[(size_t)k * N_VOX + r0];
            m0 = mask[(size_t)k * N_VOX + r0] != 0;
            g1 = nbr_idx[(size_t)k * N_VOX + r1];
            m1 = mask[(size_t)k * N_VOX + r1] != 0;
        }
        const float* src0 = feats + (size_t)g0 * CIN + lhalf * 2;
        const float* src1 = feats + (size_t)g1 * CIN + lhalf * 2;

        // ---- 16 K-chunks of 4, 4 column tiles of 16, 2 row tiles ----
        #pragma unroll
        for (int t = 0; t < 16; ++t) {
            v2f fa, fb;
            if (m0) { fa = *(const v2f*)(src0 + 4 * t); }
            else    { fa.x = 0.0f; fa.y = 0.0f; }
            if (m1) { fb = *(const v2f*)(src1 + 4 * t); }
            else    { fb.x = 0.0f; fb.y = 0.0f; }

            const int cin0 = 4 * t + 2 * lhalf;       // B rows for this lane
            const float* wp = &Wlds[cin0 * WPAD + lrow];

            v2f b;
            b.x = wp[0];        b.y = wp[WPAD + 0];
            a00 = __builtin_amdgcn_wmma_f32_16x16x4_f32(false, fa, false, b,
                                                        (short)0, a00, false, false);
            a10 = __builtin_amdgcn_wmma_f32_16x16x4_f32(false, fb, false, b,
                                                        (short)0, a10, false, false);
            b.x = wp[16];       b.y = wp[WPAD + 16];
            a01 = __builtin_amdgcn_wmma_f32_16x16x4_f32(false, fa, false, b,
                                                        (short)0, a01, false, false);
            a11 = __builtin_amdgcn_wmma_f32_16x16x4_f32(false, fb, false, b,
                                                        (short)0, a11, false, false);
            b.x = wp[32];       b.y = wp[WPAD + 32];
            a02 = __builtin_amdgcn_wmma_f32_16x16x4_f32(false, fa, false, b,
                                                        (short)0, a02, false, false);
            a12 = __builtin_amdgcn_wmma_f32_16x16x4_f32(false, fb, false, b,
                                                        (short)0, a12, false, false);
            b.x = wp[48];       b.y = wp[WPAD + 48];
            a03 = __builtin_amdgcn_wmma_f32_16x16x4_f32(false, fa, false, b,
                                                        (short)0, a03, false, false);
            a13 = __builtin_amdgcn_wmma_f32_16x16x4_f32(false, fb, false, b,
                                                        (short)0, a13, false, false);
        }
    }

    // ---- store conv tiles (C/D layout: VGPR j -> rows j / j+8) ----
    if (valid) {
        #pragma unroll
        for (int j = 0; j < 8; ++j) {
            int row  = row0 + j + 8 * lhalf;
            float* o = conv_out + (size_t)row * COUT + lrow;
            o[0]  = a00[j];
            o[16] = a01[j];
            o[32] = a02[j];
            o[48] = a03[j];
        }
        #pragma unroll
        for (int j = 0; j < 8; ++j) {
            int row  = row0 + 16 + j + 8 * lhalf;
            float* o = conv_out + (size_t)row * COUT + lrow;
            o[0]  = a10[j];
            o[16] = a11[j];
            o[32] = a12[j];
            o[48] = a13[j];
        }
    }

    // ---- deterministic per-block column sum / sumsq reduction ----
    float s0 = 0.f, s1 = 0.f, s2 = 0.f, s3 = 0.f;
    float q0 = 0.f, q1 = 0.f, q2 = 0.f, q3 = 0.f;
    #pragma unroll
    for (int j = 0; j < 8; ++j) {
        s0 += a00[j] + a10[j]; q0 += a00[j] * a00[j] + a10[j] * a10[j];
        s1 += a01[j] + a11[j]; q1 += a01[j] * a01[j] + a11[j] * a11[j];
        s2 += a02[j] + a12[j]; q2 += a02[j] * a02[j] + a12[j] * a12[j];
        s3 += a03[j] + a13[j]; q3 += a03[j] * a03[j] + a13[j] * a13[j];
    }
    // fold lane l+16 into lane l (same column, other row half)
    s0 += __shfl_xor(s0, 16, 32);  q0 += __shfl_xor(q0, 16, 32);
    s1 += __shfl_xor(s1, 16, 32);  q1 += __shfl_xor(q1, 16, 32);
    s2 += __shfl_xor(s2, 16, 32);  q2 += __shfl_xor(q2, 16, 32);
    s3 += __shfl_xor(s3, 16, 32);  q3 += __shfl_xor(q3, 16, 32);

    if (lane < 16) {
        float* rw = &red[wave * 128];
        rw[ 0 + lrow] = s0;  rw[64 +  0 + lrow] = q0;
        rw[16 + lrow] = s1;  rw[64 + 16 + lrow] = q1;
        rw[32 + lrow] = s2;  rw[64 + 32 + lrow] = q2;
        rw[48 + lrow] = s3;  rw[64 + 48 + lrow] = q3;
    }
    __syncthreads();
    if (tid < 128) {
        float tot = 0.f;
        #pragma unroll
        for (int w = 0; w < WAVES_PER_BLOCK; ++w)
            tot += red[w * 128 + tid];
        part[(size_t)blockIdx.x * 128 + tid] = tot;
    }
}

// ---------------------------------------------------------------------------
// Kernel 2: fold per-block partials into mean[64] / rstd[64] (fixed order).
// ---------------------------------------------------------------------------
__global__ void stats_kernel(const float* __restrict__ part, int nblocks,
                             float* __restrict__ stats)
{
    int c = threadIdx.x;
    if (c >= 64) return;
    float s = 0.f, q = 0.f;
    for (int b = 0; b < nblocks; ++b) {
        s += part[(size_t)b * 128 + c];
        q += part[(size_t)b * 128 + 64 + c];
    }
    float inv_n = 1.0f / (float)N_VOX;
    float mean  = s * inv_n;
    float var   = q * inv_n - mean * mean;
    stats[c]      = mean;
    stats[64 + c] = rsqrtf(var + BN_EPS);
}

// ---------------------------------------------------------------------------
// Kernel 3: in-place BN + ReLU over d_out.
// ---------------------------------------------------------------------------
__global__ __launch_bounds__(256) void bnrelu_kernel(
    float* __restrict__ out, const float* __restrict__ stats,
    const float* __restrict__ gamma, const float* __restrict__ beta)
{
    size_t gid = (size_t)blockIdx.x * 256 + threadIdx.x;   // N*64 total
    int c = (int)(gid & 63);
    float v = out[gid];
    float y = (v - stats[c]) * stats[64 + c] * gamma[c] + beta[c];
    out[gid] = fmaxf(y, 0.0f);
}

// ---------------------------------------------------------------------------
extern "C" void kernel_launch(void* const* d_in, const int* in_sizes, int n_in,
                              void* d_out, int out_size, void* d_ws, size_t ws_size,
                              hipStream_t stream)
{
    const float* feats   = (const float*)d_in[0];   // [N,64]
    const float* W       = (const float*)d_in[1];   // [27,64,64]
    const float* gamma   = (const float*)d_in[2];   // [64]
    const float* beta    = (const float*)d_in[3];   // [64]
    const int*   nbr_idx = (const int*)  d_in[4];   // [27,N]
    const int*   mask    = (const int*)  d_in[5];   // [27,N]
    float* out = (float*)d_out;                     // [N,64]

    const int nblocks = (N_VOX + ROWS_PER_BLOCK - 1) / ROWS_PER_BLOCK;   // 782
    float* part  = (float*)d_ws;                    // [nblocks,128]
    float* stats = part + (size_t)nblocks * 128;    // mean[64] + rstd[64]

    conv_wmma_kernel<<<nblocks, 256, 0, stream>>>(feats, W, nbr_idx, mask, out, part);
    stats_kernel<<<1, 64, 0, stream>>>(part, nblocks, stats);
    const int total = N_VOX * COUT;                 // 12.8M
    bnrelu_kernel<<<total / 256, 256, 0, stream>>>(out, stats, gamma, beta);
}